// Net_43851616092221
// MI455X (gfx1250) — compile-verified
//
#include <hip/hip_runtime.h>
#include <hip/hip_bf16.h>

typedef __attribute__((ext_vector_type(2))) float v2f;
typedef __attribute__((ext_vector_type(8))) float v8f;

#define HID 128
#define COS_EPS 1e-8f

// ---------------- zero ----------------
__global__ void k_zero(float* __restrict__ p, int n) {
    int i = blockIdx.x * blockDim.x + threadIdx.x;
    if (i < n) p[i] = 0.0f;
}

// ---------------- degrees ----------------
__global__ void k_degrees(const int* __restrict__ src, const int* __restrict__ dst,
                          float* __restrict__ degout, float* __restrict__ degin, int E) {
    int e = blockIdx.x * blockDim.x + threadIdx.x;
    if (e >= E) return;
    atomicAdd(&degout[src[e]], 1.0f);
    atomicAdd(&degin[dst[e]], 1.0f);
}

// deg -> rsqrt(max(deg,1)) in place (both arrays)
__global__ void k_degfix(float* __restrict__ degout, float* __restrict__ degin, int N) {
    int i = blockIdx.x * blockDim.x + threadIdx.x;
    if (i >= N) return;
    degout[i] = rsqrtf(fmaxf(degout[i], 1.0f));
    degin[i]  = rsqrtf(fmaxf(degin[i], 1.0f));
}

// ---------------- layer 1 scatter (2 features) ----------------
__global__ void k_scatter1(const int* __restrict__ src, const int* __restrict__ dst,
                           const float* __restrict__ pos, const float* __restrict__ rs_out,
                           float* __restrict__ agg1, int E) {
    int e = blockIdx.x * blockDim.x + threadIdx.x;
    if (e >= E) return;
    int s = src[e], d = dst[e];
    float rs = rs_out[s];
    atomicAdd(&agg1[d * 2 + 0], pos[s * 2 + 0] * rs);
    atomicAdd(&agg1[d * 2 + 1], pos[s * 2 + 1] * rs);
}

// ---------------- layer 1 dense: h1n = ((agg1 * rs_in) @ W1 + b1) * rs_out ----------------
__global__ void k_node1(const float* __restrict__ agg1, const float* __restrict__ rs_in,
                        const float* __restrict__ rs_out,
                        const float* __restrict__ W1, const float* __restrict__ b1,
                        float* __restrict__ h1n, int N) {
    int idx = blockIdx.x * blockDim.x + threadIdx.x;
    int n = idx >> 7, f = idx & 127;
    if (n >= N) return;
    float ri = rs_in[n];
    float a0 = agg1[n * 2 + 0] * ri;
    float a1 = agg1[n * 2 + 1] * ri;
    float h = fmaf(a0, W1[f], fmaf(a1, W1[HID + f], b1[f]));
    h1n[idx] = h * rs_out[n];
}

// ---------------- layer 2 scatter: agg2[dst] += h1n[src] (128 features) ----------------
__global__ void __launch_bounds__(128)
k_scatter2(const int* __restrict__ src, const int* __restrict__ dst,
           const float* __restrict__ h1n, float* __restrict__ agg2, int E) {
    int e = blockIdx.x;
    if (e >= E) return;
    int s = src[e], d = dst[e];
    int f = threadIdx.x;
    atomicAdd(&agg2[d * HID + f], h1n[s * HID + f]);
}

// ---------------- layer 2 GEMM via fp32 WMMA: h2 = (agg2*rs_in) @ W2 + b2 ----------------
// One wave32 per 16x16 output tile. K loop in steps of 4 (V_WMMA_F32_16X16X4_F32).
__global__ void __launch_bounds__(256)
k_gemm2_wmma(const float* __restrict__ agg2, const float* __restrict__ rs_in,
             const float* __restrict__ W2, const float* __restrict__ b2,
             float* __restrict__ h2, int N) {
    int gid  = blockIdx.x * blockDim.x + threadIdx.x;
    int wave = gid >> 5;
    int lane = threadIdx.x & 31;
    int numTiles = (N >> 4) * (HID >> 4);
    if (wave >= numTiles) return;           // wave-uniform: EXEC stays all-1s for WMMA

    int tm = wave >> 3;                     // row tile (HID/16 == 8 col tiles)
    int tn = wave & 7;
    int row0 = tm << 4;
    int col0 = tn << 4;

    int m  = lane & 15;                     // row (A) / col (B,D) within tile
    int kb = (lane >> 4) << 1;              // lanes 0-15 -> K {0,1}; lanes 16-31 -> K {2,3}

    int arow = row0 + m;
    float rs = rs_in[arow];
    const float* Arow = agg2 + (size_t)arow * HID;

    v8f acc = {};
    #pragma unroll
    for (int k0 = 0; k0 < HID; k0 += 4) {
        // A 16x4 fp32 tile, ISA layout
        v2f a;
        {
            const float2 av = *(const float2*)(Arow + k0 + kb);   // even offset -> 8B aligned
            a.x = av.x * rs;
            a.y = av.y * rs;
        }
        // B 4x16 fp32 tile (rows of W2)
        v2f b;
        b.x = W2[(k0 + kb + 0) * HID + col0 + m];
        b.y = W2[(k0 + kb + 1) * HID + col0 + m];
        acc = __builtin_amdgcn_wmma_f32_16x16x4_f32(
                  false, a, false, b, (short)0, acc, false, false);
    }

    float bias = b2[col0 + m];
    #pragma unroll
    for (int v = 0; v < 8; ++v) {
        int r = row0 + v + ((lane >> 4) << 3);   // D layout: row = vgpr + 8*(lane>=16)
        h2[(size_t)r * HID + col0 + m] = acc[v] + bias;
    }
}

// ---------------- per-node inverse norm (one wave per node) ----------------
__global__ void __launch_bounds__(256)
k_invnorm(const float* __restrict__ h2, float* __restrict__ inv_n, int N) {
    int gid  = blockIdx.x * blockDim.x + threadIdx.x;
    int node = gid >> 5;
    int lane = threadIdx.x & 31;
    if (node >= N) return;
    const float4* p = (const float4*)(h2 + (size_t)node * HID);
    float4 v = p[lane];
    float s = v.x * v.x + v.y * v.y + v.z * v.z + v.w * v.w;
    #pragma unroll
    for (int off = 16; off > 0; off >>= 1) s += __shfl_down(s, off, 32);
    if (lane == 0) inv_n[node] = 1.0f / fmaxf(sqrtf(s), COS_EPS);
}

// ---------------- edge cosine similarity (one wave per edge) ----------------
__global__ void __launch_bounds__(256)
k_cosine(const int* __restrict__ src, const int* __restrict__ dst,
         const float* __restrict__ h2, const float* __restrict__ inv_n,
         float* __restrict__ out, int E) {
    int gid  = blockIdx.x * blockDim.x + threadIdx.x;
    int e    = gid >> 5;
    int lane = threadIdx.x & 31;
    if (e >= E) return;
    int s = src[e], d = dst[e];
    const float4* ps = (const float4*)(h2 + (size_t)s * HID);
    const float4* pd = (const float4*)(h2 + (size_t)d * HID);
    float4 a = ps[lane];
    float4 b = pd[lane];
    float acc = a.x * b.x + a.y * b.y + a.z * b.z + a.w * b.w;
    #pragma unroll
    for (int off = 16; off > 0; off >>= 1) acc += __shfl_down(acc, off, 32);
    if (lane == 0) out[e] = acc * inv_n[s] * inv_n[d];
}

extern "C" void kernel_launch(void* const* d_in, const int* in_sizes, int n_in,
                              void* d_out, int out_size, void* d_ws, size_t ws_size,
                              hipStream_t stream) {
    const float* positions = (const float*)d_in[0];
    const int*   src       = (const int*)d_in[1];
    const int*   dst       = (const int*)d_in[2];
    const float* W1        = (const float*)d_in[3];
    const float* b1        = (const float*)d_in[4];
    const float* W2        = (const float*)d_in[5];
    const float* b2        = (const float*)d_in[6];
    float*       out       = (float*)d_out;

    const int N = in_sizes[0] / 2;
    const int E = in_sizes[1];

    // workspace layout (floats)
    float* ws     = (float*)d_ws;
    float* degout = ws;                 // N   (becomes rs_out)
    float* degin  = degout + N;         // N   (becomes rs_in)
    float* agg1   = degin + N;          // 2N
    float* inv_n  = agg1 + 2 * N;       // N
    float* h1n    = inv_n + N;          // N*HID
    float* agg2   = h1n + (size_t)N * HID; // N*HID
    float* h2     = h1n;                // reuse: h1n dead after scatter2

    const int TB = 256;
    auto blocks = [](long long n, int tb) { return (int)((n + tb - 1) / tb); };

    // zero accumulators (deg/agg1 contiguous 4N, plus agg2)
    k_zero<<<blocks(4LL * N, TB), TB, 0, stream>>>(degout, 4 * N);
    k_zero<<<blocks((long long)N * HID, TB), TB, 0, stream>>>(agg2, N * HID);

    // degrees -> rsqrt factors
    k_degrees<<<blocks(E, TB), TB, 0, stream>>>(src, dst, degout, degin, E);
    k_degfix<<<blocks(N, TB), TB, 0, stream>>>(degout, degin, N);

    // layer 1
    k_scatter1<<<blocks(E, TB), TB, 0, stream>>>(src, dst, positions, degout, agg1, E);
    k_node1<<<blocks((long long)N * HID, TB), TB, 0, stream>>>(agg1, degin, degout, W1, b1, h1n, N);

    // layer 2
    k_scatter2<<<E, HID, 0, stream>>>(src, dst, h1n, agg2, E);
    int numTiles = (N / 16) * (HID / 16);
    k_gemm2_wmma<<<blocks((long long)numTiles * 32, TB), TB, 0, stream>>>(agg2, degin, W2, b2, h2, N);

    // cosine similarity
    k_invnorm<<<blocks((long long)N * 32, TB), TB, 0, stream>>>(h2, inv_n, N);
    k_cosine<<<blocks((long long)E * 32, TB), TB, 0, stream>>>(src, dst, h2, inv_n, out, E);
}